// GroupSorter_14972255994388
// MI455X (gfx1250) — compile-verified
//
#include <hip/hip_runtime.h>

// GroupSorter: B=512 groups, N=64 rows/group, C=1024 features, fp32.
// out = concat(out_sorted [B, N*C], out_input [B, N*C]) flat fp32.

typedef float v2f __attribute__((ext_vector_type(2)));
typedef float v4f __attribute__((ext_vector_type(4)));
typedef float v8f __attribute__((ext_vector_type(8)));
typedef int   v4i __attribute__((ext_vector_type(4)));

#define BGRP 512
#define NROW 64
#define CDIM 1024
#define KCH  128           // K-chunk width held in LDS
#define LSTR 132           // KCH + 4 pad floats -> rows 4 banks apart (conflict-free frag loads)
#define NTHR 512           // 16 waves: wave (tm,tn) owns 16x16 tile (tm,tn) of the Gram matrix

#if __has_builtin(__builtin_amdgcn_global_load_async_to_lds_b128)
#define USE_ASYNC 1
#else
#define USE_ASYNC 0
#endif

__global__ __launch_bounds__(NTHR) void grs_kernel(const float* __restrict__ feats,
                                                   float* __restrict__ out_sorted,
                                                   float* __restrict__ out_input) {
    __shared__ float gs[NROW * LSTR];     // 33 KB chunk tile
    __shared__ float inv[NROW];           // 1/max(||row||, eps)
    __shared__ float relpart[4][NROW];    // per-tn partial row sums
    __shared__ float rel[NROW];
    __shared__ int   order[NROW];

    const int b    = blockIdx.x;
    const int t    = threadIdx.x;
    const int wave = t >> 5;
    const int lane = t & 31;
    const float* gbase = feats + (size_t)b * NROW * CDIM;

    // ---- Phase 1: raw Gram S = g * g^T via V_WMMA_F32_16X16X4_F32, with the
    //      row sum-of-squares folded into the same LDS-resident chunks. ----
    const int tm   = wave >> 2;
    const int tn   = wave & 3;
    const int mrow = tm * 16 + (lane & 15);     // A rows (M = lane%16)
    const int nrow = tn * 16 + (lane & 15);     // B "rows" (N = lane%16)
    const int koff = (lane >> 4) * 2;           // VGPR0/1 hold K = koff, koff+1
    v8f acc = {};
    float ss[4] = {0.0f, 0.0f, 0.0f, 0.0f};     // norm partials: rows wave*4 .. wave*4+3

    for (int kc = 0; kc < CDIM; kc += KCH) {
        // stage 64 x 128 fp32 chunk into LDS
#if USE_ASYNC
        #pragma unroll
        for (int i = 0; i < 16; ++i) {
            const int f   = t + i * NTHR;       // float4 index 0..8191
            const int row = f >> 5;             // 32 float4 per row
            const int c4  = (f & 31) * 4;
            __builtin_amdgcn_global_load_async_to_lds_b128(
                (v4i*)(gbase + row * CDIM + kc + c4),
                (v4i*)(&gs[row * LSTR + c4]), 0, 0);
        }
        #if __has_builtin(__builtin_amdgcn_s_wait_asynccnt)
        __builtin_amdgcn_s_wait_asynccnt(0);
        #else
        asm volatile("s_wait_asynccnt 0x0" ::: "memory");
        #endif
        __syncthreads();
#else
        #pragma unroll
        for (int i = 0; i < 16; ++i) {
            const int f   = t + i * NTHR;       // float4 index 0..8191
            const int row = f >> 5;             // 32 float4 per row
            const int c4  = (f & 31) * 4;
            v4f v = *(const v4f*)(gbase + row * CDIM + kc + c4);
            *(v4f*)(&gs[row * LSTR + c4]) = v;
        }
        __syncthreads();
#endif

        const float* ap = &gs[mrow * LSTR + koff];
        const float* bp = &gs[nrow * LSTR + koff];
        #pragma unroll
        for (int ks = 0; ks < KCH; ks += 4) {
            v2f a  = *(const v2f*)(ap + ks);
            v2f bb = *(const v2f*)(bp + ks);
            // D = A(16x4) * B(4x16) + C, fp32
            acc = __builtin_amdgcn_wmma_f32_16x16x4_f32(
                false, a, false, bb, (short)0, acc, false, false);
        }

        // norm partials from the same LDS chunk (wave owns rows wave*4..+3)
        #pragma unroll
        for (int i = 0; i < 4; ++i) {
            const int row = wave * 4 + i;
            v4f v = *(const v4f*)(&gs[row * LSTR + lane * 4]);
            ss[i] = fmaf(v.x, v.x, ss[i]); ss[i] = fmaf(v.y, v.y, ss[i]);
            ss[i] = fmaf(v.z, v.z, ss[i]); ss[i] = fmaf(v.w, v.w, ss[i]);
        }
        __syncthreads();
    }

    // finalize inverse norms
    #pragma unroll
    for (int i = 0; i < 4; ++i) {
        float s = ss[i];
        #pragma unroll
        for (int off = 16; off; off >>= 1) s += __shfl_xor(s, off, 32);
        if (lane == 0) inv[wave * 4 + i] = 1.0f / fmaxf(sqrtf(s), 1e-12f);
    }
    __syncthreads();

    // ---- Phase 2: rel[n] = inv[n]/64 * sum_m S[n,m]*inv[m] (deterministic reduce) ----
    {
        const float invn = inv[nrow];
        float w[8];
        #pragma unroll
        for (int j = 0; j < 8; ++j) w[j] = acc[j] * invn;
        #pragma unroll
        for (int off = 1; off < 16; off <<= 1) {
            #pragma unroll
            for (int j = 0; j < 8; ++j) w[j] += __shfl_xor(w[j], off, 32);
        }
        if ((lane & 15) == 0) {
            const int mb = tm * 16 + (lane >> 4) * 8;   // C/D layout: VGPR j -> M = 8*(lane/16)+j
            #pragma unroll
            for (int j = 0; j < 8; ++j) relpart[tn][mb + j] = w[j];
        }
    }
    __syncthreads();

    if (t < NROW) {
        const float s = relpart[0][t] + relpart[1][t] + relpart[2][t] + relpart[3][t];
        rel[t] = s * inv[t] * (1.0f / 64.0f);
    }
    __syncthreads();

    // ---- Phase 3: stable descending rank == jnp.argsort(-rel) ----
    if (t < NROW) {
        const float mine = rel[t];
        int r = 0;
        for (int m = 0; m < NROW; ++m) {
            const float o = rel[m];
            r += (o > mine) || (o == mine && m < t);
        }
        order[r] = t;
    }
    __syncthreads();

    // ---- Phase 4: outputs. Reads are L2-hot; writes are never re-read -> NT stores
    //      so the 268 MB output stream doesn't evict feats from the 192 MB L2. ----
    float* outs = out_sorted + (size_t)b * NROW * CDIM;
    float* outi = out_input  + (size_t)b * NROW * CDIM;
    #pragma unroll 4
    for (int i = 0; i < 32; ++i) {
        const int f   = t + i * NTHR;           // float4 index 0..16383
        const int row = f >> 8;                 // 256 float4 per row
        const int c4  = (f & 255) * 4;
        v4f v = *(const v4f*)(gbase + row * CDIM + c4);
        __builtin_nontemporal_store(v, (v4f*)(outi + row * CDIM + c4));
        const int sr = order[row];
        v4f u = *(const v4f*)(gbase + sr * CDIM + c4);
        __builtin_nontemporal_store(u, (v4f*)(outs + row * CDIM + c4));
    }
}

extern "C" void kernel_launch(void* const* d_in, const int* in_sizes, int n_in,
                              void* d_out, int out_size, void* d_ws, size_t ws_size,
                              hipStream_t stream) {
    (void)in_sizes; (void)n_in; (void)d_ws; (void)ws_size; (void)out_size;
    const float* feats = (const float*)d_in[0];
    // d_in[1] (labels) encodes contiguous uniform groups -> static layout, unused.
    float* out        = (float*)d_out;
    float* out_sorted = out;                                  // [B, N*C]
    float* out_input  = out + (size_t)BGRP * NROW * CDIM;     // [B, N*C]
    grs_kernel<<<BGRP, NTHR, 0, stream>>>(feats, out_sorted, out_input);
}